// PolyGCL_71184787964540
// MI455X (gfx1250) — compile-verified
//
#include <hip/hip_runtime.h>

// ---------------- problem constants (match reference) ----------------
constexpr int NN   = 50000;    // nodes
constexpr int EE   = 800000;   // edges
constexpr int INF  = 500;      // input features
constexpr int HIDF = 128;      // hidden
constexpr int OUTF = 128;      // output
constexpr int KP   = 10;       // polynomial order
constexpr long long NH = (long long)NN * HIDF;   // 6.4M

typedef __attribute__((ext_vector_type(2))) float v2f;
typedef __attribute__((ext_vector_type(8))) float v8f;

// ---------------- degree / norm ----------------
__global__ void deg_kernel(const int* __restrict__ dst, float* __restrict__ deg, int E) {
    int e = blockIdx.x * blockDim.x + threadIdx.x;
    if (e < E) unsafeAtomicAdd(&deg[dst[e]], 1.0f);
}

__global__ void dinv_kernel(float* __restrict__ deg, int n) {
    int i = blockIdx.x * blockDim.x + threadIdx.x;
    if (i < n) deg[i] = rsqrtf(fmaxf(deg[i], 1.0f));   // in place: deg -> d^{-1/2}
}

__global__ void norm_kernel(const int* __restrict__ src, const int* __restrict__ dst,
                            const float* __restrict__ dinv, float* __restrict__ nrm, int E) {
    int e = blockIdx.x * blockDim.x + threadIdx.x;
    if (e < E) nrm[e] = dinv[src[e]] * dinv[dst[e]];
}

// ---------------- fp32 WMMA GEMM: D = A[MxK] * B[KxN] + bias (optional relu) ----
// Row-major, M%16==0, K%4==0, N%16==0. One wave per 16x16 output tile.
// A-frag (16x4 f32): lanes 0-15 -> K = k0+{0,1}; lanes 16-31 -> K = k0+{2,3}; row = lane&15
// B-frag (4x16 f32): mirrored (lanes 0-15 K=k0+{0,1}, lanes 16-31 K=k0+{2,3}); col = lane&15
// C/D  (16x16 f32): vgpr r -> row = r + 8*(lane>>4), col = lane&15
__global__ void __launch_bounds__(256)
wmma_gemm_f32(const float* __restrict__ A, const float* __restrict__ B,
              const float* __restrict__ bias, float* __restrict__ D,
              int M, int K, int N, int relu)
{
    const int lane  = threadIdx.x & 31;
    const int wave  = threadIdx.x >> 5;
    const int tilesN = N >> 4;
    const int tile  = blockIdx.x * 8 + wave;
    const int tM = tile / tilesN;
    const int tN = tile % tilesN;
    if (tM * 16 >= M) return;                 // wave-uniform (EXEC stays all-ones)

    const int half = lane >> 4;               // 0: K pair {0,1}, 1: K pair {2,3}
    const int mn   = lane & 15;               // A row within tile / B-D col within tile

    const float* Arow = A + (long long)(tM * 16 + mn) * K;
    const float* Bcol = B + (tN * 16 + mn);

    v8f c = {};
    for (int k0 = 0; k0 < K; k0 += 4) {
        const int ka = k0 + 2 * half;
        v2f a, b;
        a.x = Arow[ka];
        a.y = Arow[ka + 1];
        b.x = Bcol[(long long)ka * N];
        b.y = Bcol[(long long)(ka + 1) * N];
        c = __builtin_amdgcn_wmma_f32_16x16x4_f32(false, a, false, b,
                                                  (short)0, c, false, false);
    }

    const int col = tN * 16 + mn;
    const float bv = bias ? bias[col] : 0.0f;
#pragma unroll
    for (int r = 0; r < 8; ++r) {
        const int row = tM * 16 + r + 8 * half;
        float v = c[r] + bv;
        if (relu) v = fmaxf(v, 0.0f);
        D[(long long)row * N + col] = v;
    }
}

// ---------------- elementwise helpers (float4 over N*HID) ----------------
__global__ void init_kernel(const float* __restrict__ h, float* __restrict__ t,
                            float* __restrict__ acc, int n4) {
    int i = blockIdx.x * blockDim.x + threadIdx.x;
    if (i < n4) {
        ((float4*)t)[i]   = ((const float4*)h)[i];
        ((float4*)acc)[i] = make_float4(0.f, 0.f, 0.f, 0.f);
    }
}

__global__ void axpy_kernel(const float* __restrict__ gamma, int k,
                            const float* __restrict__ t, float* __restrict__ acc, int n4) {
    int i = blockIdx.x * blockDim.x + threadIdx.x;
    if (i < n4) {
        const float g = gamma[k];
        float4 tv = ((const float4*)t)[i];
        float4 av = ((float4*)acc)[i];
        av.x += g * tv.x; av.y += g * tv.y; av.z += g * tv.z; av.w += g * tv.w;
        ((float4*)acc)[i] = av;
    }
}

__global__ void update_kernel(float* __restrict__ t, const float* __restrict__ pt,
                              int high, int n4) {
    int i = blockIdx.x * blockDim.x + threadIdx.x;
    if (i < n4) {
        float4 pv = ((const float4*)pt)[i];
        if (high) {
            float4 tv = ((float4*)t)[i];
            tv.x -= pv.x; tv.y -= pv.y; tv.z -= pv.z; tv.w -= pv.w;
            ((float4*)t)[i] = tv;
        } else {
            ((float4*)t)[i] = pv;
        }
    }
}

// ---------------- sparse propagation: pt[dst] += norm * t[src] ----------------
// 32 lanes per edge, 4 features (float4) per lane.
__global__ void scatter_kernel(const int* __restrict__ src, const int* __restrict__ dst,
                               const float* __restrict__ nrm, const float* __restrict__ t,
                               float* __restrict__ pt, int E)
{
    long long gid = (long long)blockIdx.x * blockDim.x + threadIdx.x;
    int e = (int)(gid >> 5);
    if (e >= E) return;
    int c = ((int)gid & 31) * 4;
    const int   s = src[e];
    const int   d = dst[e];
    const float w = nrm[e];
    float4 v = *(const float4*)(t + (long long)s * HIDF + c);
    float* o = pt + (long long)d * HIDF + c;
    unsafeAtomicAdd(o + 0, w * v.x);
    unsafeAtomicAdd(o + 1, w * v.y);
    unsafeAtomicAdd(o + 2, w * v.z);
    unsafeAtomicAdd(o + 3, w * v.w);
}

// ---------------- batch-norm statistics ----------------
__global__ void bn_stats_kernel(const float* __restrict__ acc,
                                float* __restrict__ sums, float* __restrict__ sqs, int n)
{
    __shared__ float s_sum[HIDF];
    __shared__ float s_sq[HIDF];
    const int tid = threadIdx.x;
    if (tid < HIDF) { s_sum[tid] = 0.f; s_sq[tid] = 0.f; }
    __syncthreads();

    const int c = (tid & 31) * 4;                       // feature chunk
    const int rows_per_blk = blockDim.x >> 5;           // 8
    int row = blockIdx.x * rows_per_blk + (tid >> 5);
    const int rstride = gridDim.x * rows_per_blk;
    float4 s = make_float4(0.f, 0.f, 0.f, 0.f);
    float4 q = make_float4(0.f, 0.f, 0.f, 0.f);
    for (; row < n; row += rstride) {
        float4 v = *(const float4*)(acc + (long long)row * HIDF + c);
        s.x += v.x; s.y += v.y; s.z += v.z; s.w += v.w;
        q.x += v.x * v.x; q.y += v.y * v.y; q.z += v.z * v.z; q.w += v.w * v.w;
    }
    atomicAdd(&s_sum[c + 0], s.x); atomicAdd(&s_sum[c + 1], s.y);
    atomicAdd(&s_sum[c + 2], s.z); atomicAdd(&s_sum[c + 3], s.w);
    atomicAdd(&s_sq[c + 0], q.x);  atomicAdd(&s_sq[c + 1], q.y);
    atomicAdd(&s_sq[c + 2], q.z);  atomicAdd(&s_sq[c + 3], q.w);
    __syncthreads();
    if (tid < HIDF) {
        unsafeAtomicAdd(&sums[tid], s_sum[tid]);
        unsafeAtomicAdd(&sqs[tid],  s_sq[tid]);
    }
}

__global__ void bn_finalize_kernel(const float* __restrict__ sums, const float* __restrict__ sqs,
                                   const float* __restrict__ bn_scale, const float* __restrict__ bn_shift,
                                   float* __restrict__ aS, float* __restrict__ bS, int n)
{
    int f = threadIdx.x;
    if (f < HIDF) {
        float mean = sums[f] / (float)n;
        float var  = sqs[f] / (float)n - mean * mean;
        float inv  = rsqrtf(var + 1e-5f);
        float a = bn_scale[f] * inv;
        aS[f] = a;
        bS[f] = bn_shift[f] - mean * a;
    }
}

__global__ void bn_apply_kernel(const float* __restrict__ acc, const float* __restrict__ aS,
                                const float* __restrict__ bS, float* __restrict__ hn, int n4)
{
    int i = blockIdx.x * blockDim.x + threadIdx.x;
    if (i < n4) {
        int c = (i & 31) * 4;                           // 32 float4 chunks per 128-wide row
        float4 v = ((const float4*)acc)[i];
        float4 a = *(const float4*)(aS + c);
        float4 b = *(const float4*)(bS + c);
        v.x = v.x * a.x + b.x; v.y = v.y * a.y + b.y;
        v.z = v.z * a.z + b.z; v.w = v.w * a.w + b.w;
        ((float4*)hn)[i] = v;
    }
}

// ---------------- host orchestration ----------------
extern "C" void kernel_launch(void* const* d_in, const int* in_sizes, int n_in,
                              void* d_out, int out_size, void* d_ws, size_t ws_size,
                              hipStream_t stream)
{
    (void)in_sizes; (void)n_in; (void)out_size; (void)ws_size;

    const float* x    = (const float*)d_in[0];
    const int*   ei   = (const int*)d_in[1];
    const int*   srcI = ei;
    const int*   dstI = ei + EE;
    const float* W_in = (const float*)d_in[2];
    const float* b_in = (const float*)d_in[3];
    const float* gL   = (const float*)d_in[4];
    const float* gH   = (const float*)d_in[5];
    const float* bnS  = (const float*)d_in[6];
    const float* bnB  = (const float*)d_in[7];
    const float* W_up = (const float*)d_in[8];
    const float* b_up = (const float*)d_in[9];
    float* out = (float*)d_out;

    // workspace layout (floats)
    float* ws   = (float*)d_ws;
    float* dinv = ws;               // N   (degree -> d^{-1/2} in place)
    float* nrm  = dinv + NN;        // E
    float* h    = nrm + EE;         // N*HID
    float* t    = h + NH;           // N*HID
    float* pt   = t + NH;           // N*HID
    float* acc  = pt + NH;          // N*HID
    float* sums = acc + NH;         // HID
    float* sqs  = sums + HIDF;      // HID
    float* aS   = sqs + HIDF;       // HID
    float* bS   = aS + HIDF;        // HID

    const int eb  = (EE + 255) / 256;
    const int nb  = (NN + 255) / 256;
    const int n4  = (int)(NH / 4);
    const int e4b = (n4 + 255) / 256;
    const int sb  = (int)(((long long)EE * 32 + 255) / 256);

    // ---- graph norm ----
    hipMemsetAsync(dinv, 0, NN * sizeof(float), stream);
    deg_kernel<<<eb, 256, 0, stream>>>(dstI, dinv, EE);
    dinv_kernel<<<nb, 256, 0, stream>>>(dinv, NN);
    norm_kernel<<<eb, 256, 0, stream>>>(srcI, dstI, dinv, nrm, EE);

    // ---- h = x @ W_in + b_in  (WMMA f32) ----
    {
        const int tiles = (NN / 16) * (HIDF / 16);
        wmma_gemm_f32<<<tiles / 8, 256, 0, stream>>>(x, W_in, b_in, h, NN, INF, HIDF, 0);
    }

    // ---- two polynomial branches (low-pass, high-pass) ----
    for (int enc = 0; enc < 2; ++enc) {
        const float* gamma = (enc == 0) ? gL : gH;
        const int high = enc;

        init_kernel<<<e4b, 256, 0, stream>>>(h, t, acc, n4);
        for (int k = 0; k <= KP; ++k) {
            axpy_kernel<<<e4b, 256, 0, stream>>>(gamma, k, t, acc, n4);
            if (k < KP) {
                hipMemsetAsync(pt, 0, NH * sizeof(float), stream);
                scatter_kernel<<<sb, 256, 0, stream>>>(srcI, dstI, nrm, t, pt, EE);
                update_kernel<<<e4b, 256, 0, stream>>>(t, pt, high, n4);
            }
        }

        // batch norm (training-mode statistics over nodes)
        hipMemsetAsync(sums, 0, 2 * HIDF * sizeof(float), stream);
        bn_stats_kernel<<<512, 256, 0, stream>>>(acc, sums, sqs, NN);
        bn_finalize_kernel<<<1, 128, 0, stream>>>(sums, sqs, bnS, bnB, aS, bS, NN);
        bn_apply_kernel<<<e4b, 256, 0, stream>>>(acc, aS, bS, pt, n4);   // pt <- hn

        // z = relu(hn @ W_up + b_up) straight into output slot
        {
            const int tiles = (NN / 16) * (OUTF / 16);
            float* Dst = out + (long long)enc * NN * OUTF;
            wmma_gemm_f32<<<tiles / 8, 256, 0, stream>>>(pt, W_up, b_up, Dst, NN, HIDF, OUTF, 1);
        }
    }
}